// GCNLayer_85925115724063
// MI455X (gfx1250) — compile-verified
//
#include <hip/hip_runtime.h>

// COO SpMM: out[row[e],:] += vals[e] * embeds[col[e],:], D_FEAT = 128 (fp32).
// MI455X: wave32, 192MB L2 (embeds 51.2MB + out 51.2MB both L2-resident),
// async global->LDS staging of edge lists, L2-side fp32 atomics for scatter.

#define WAVE_SZ  32
#define TILE     32          // edges staged per wave-tile
#define BLOCK_SZ 256         // 8 waves / block
#define WPB      (BLOCK_SZ / WAVE_SZ)
#define NBLOCKS  2048
#define DFEAT    128         // reference D_FEAT; 32 lanes x float4 = one row

struct EdgeBuf {
  int   row[TILE];
  int   col[TILE];
  float val[TILE];
};

// Async copy 4B/lane from global (saddr + 32-bit voffset) into the LDS address in lds_addr.
__device__ __forceinline__ void async_gather_b32(unsigned lds_addr,
                                                 unsigned byte_off,
                                                 const void* sgpr_base) {
  asm volatile("global_load_async_to_lds_b32 %0, %1, %2"
               :: "v"(lds_addr), "v"(byte_off), "s"(sgpr_base)
               : "memory");
}

__device__ __forceinline__ void wait_async_le3() {
  asm volatile("s_wait_asynccnt 0x3" ::: "memory");
}
__device__ __forceinline__ void wait_async_le0() {
  asm volatile("s_wait_asynccnt 0x0" ::: "memory");
}

// Hardware fp32 atomic add (global_atomic_add_f32), never a CAS loop.
__device__ __forceinline__ void hw_atomic_add(float* p, float v) {
#if defined(__HIP_DEVICE_COMPILE__)
  unsafeAtomicAdd(p, v);
#else
  atomicAdd(p, v);
#endif
}

__device__ __forceinline__ void edge_update(const float* __restrict__ embeds,
                                            float* __restrict__ out,
                                            int r, int c, float v, int lane) {
  // One 512B embedding row per wave: lane l owns floats [4l, 4l+4).
  const float4 e =
      *reinterpret_cast<const float4*>(embeds + ((size_t)c << 7) + (lane << 2));
  float* dst = out + ((size_t)r << 7) + (lane << 2);
  hw_atomic_add(dst + 0, e.x * v);
  hw_atomic_add(dst + 1, e.y * v);
  hw_atomic_add(dst + 2, e.z * v);
  hw_atomic_add(dst + 3, e.w * v);
}

__global__ __launch_bounds__(BLOCK_SZ) void spmm_coo_kernel(
    const int* __restrict__ row, const int* __restrict__ col,
    const float* __restrict__ vals, const float* __restrict__ embeds,
    float* __restrict__ out, int n_edges) {
  __shared__ EdgeBuf buf[WPB][2];

  const int lane  = threadIdx.x & (WAVE_SZ - 1);
  const int w     = threadIdx.x >> 5;
  const int wave  = blockIdx.x * WPB + w;
  const int nWave = gridDim.x * WPB;
  const int nFull = n_edges / TILE;

  // Per-wave LDS byte addresses (double buffered), lane-strided.
  unsigned lrow[2], lcol[2], lval[2];
#pragma unroll
  for (int b = 0; b < 2; ++b) {
    lrow[b] = (unsigned)(uintptr_t)&buf[w][b].row[lane];
    lcol[b] = (unsigned)(uintptr_t)&buf[w][b].col[lane];
    lval[b] = (unsigned)(uintptr_t)&buf[w][b].val[lane];
  }

  // Prime the pipeline: stage this wave's first tile.
  int p = 0;
  if (wave < nFull) {
    unsigned off = (unsigned)((wave * TILE + lane) << 2);
    async_gather_b32(lrow[0], off, row);
    async_gather_b32(lcol[0], off, col);
    async_gather_b32(lval[0], off, vals);
  }

  for (int t = wave; t < nFull; t += nWave) {
    const int tn = t + nWave;
    if (tn < nFull) {
      // Stage next tile into the other buffer, then wait for the *current*
      // tile's 3 loads (async loads complete in order => ASYNCcnt<=3 works).
      unsigned off = (unsigned)((tn * TILE + lane) << 2);
      async_gather_b32(lrow[p ^ 1], off, row);
      async_gather_b32(lcol[p ^ 1], off, col);
      async_gather_b32(lval[p ^ 1], off, vals);
      wait_async_le3();
    } else {
      wait_async_le0();
    }

    // Pre-touch this tile's 32 gather rows (512B = 4 lines each) into L2.
    {
      const int myc = buf[w][p].col[lane];
      const char* pb = (const char*)(embeds + ((size_t)myc << 7));
      __builtin_prefetch(pb, 0, 1);
      __builtin_prefetch(pb + 128, 0, 1);
      __builtin_prefetch(pb + 256, 0, 1);
      __builtin_prefetch(pb + 384, 0, 1);
    }

    // Broadcast each staged edge from LDS; gather/scale/atomic-scatter.
#pragma unroll 4
    for (int j = 0; j < TILE; ++j) {
      const int   rj = buf[w][p].row[j];
      const int   cj = buf[w][p].col[j];
      const float vj = buf[w][p].val[j];
      edge_update(embeds, out, rj, cj, vj, lane);
    }
    p ^= 1;
  }

  // Tail: E % TILE edges, handled by wave 0 with guarded loads + shfl bcast.
  if (wave == 0) {
    const int base = nFull * TILE;
    const int rem  = n_edges - base;
    if (rem > 0) {
      int r = 0, c = 0;
      float v = 0.f;
      if (lane < rem) {
        r = row[base + lane];
        c = col[base + lane];
        v = vals[base + lane];
      }
      for (int j = 0; j < rem; ++j) {
        const int   rj = __shfl(r, j);
        const int   cj = __shfl(c, j);
        const float vj = __shfl(v, j);
        edge_update(embeds, out, rj, cj, vj, lane);
      }
    }
  }
}

extern "C" void kernel_launch(void* const* d_in, const int* in_sizes, int n_in,
                              void* d_out, int out_size, void* d_ws,
                              size_t ws_size, hipStream_t stream) {
  const int*   row    = (const int*)d_in[0];
  const int*   col    = (const int*)d_in[1];
  const float* vals   = (const float*)d_in[2];
  const float* embeds = (const float*)d_in[3];
  float*       out    = (float*)d_out;
  const int    n_edges = in_sizes[0];

  // Harness poisons d_out; atomic-scatter needs zeroed output each call.
  hipMemsetAsync(d_out, 0, (size_t)out_size * sizeof(float), stream);

  spmm_coo_kernel<<<dim3(NBLOCKS), dim3(BLOCK_SZ), 0, stream>>>(
      row, col, vals, embeds, out, n_edges);
}